// NGCF_47648367182220
// MI455X (gfx1250) — compile-verified
//
#include <hip/hip_runtime.h>

#define N_USERS 100000
#define N_ITEMS 50000
#define N_NODES (N_USERS + N_ITEMS)
#define DIM 64
#define HOPS 3
#define NNZ 4800000
#define N_TILES (N_NODES / 16)   // 9375, exact
#define LDS_STRIDE 66            // padded column stride for transposed weights

typedef float v2f __attribute__((ext_vector_type(2)));
typedef float v8f __attribute__((ext_vector_type(8)));

// ---------------------------------------------------------------------------
// ego0 = concat(user_emb, item_emb) -> out[:, 0, :]
// out layout: [N_NODES, HOPS+1, DIM], node n hop h dim d -> ((n*4)+h)*64 + d
// ---------------------------------------------------------------------------
__global__ __launch_bounds__(256) void ngcf_init(const float* __restrict__ user,
                                                 const float* __restrict__ item,
                                                 float* __restrict__ out) {
  int idx = blockIdx.x * blockDim.x + threadIdx.x;
  if (idx >= N_NODES * DIM) return;
  int node = idx >> 6;
  int d = idx & 63;
  float v = (node < N_USERS) ? user[node * DIM + d]
                             : item[(node - N_USERS) * DIM + d];
  out[(node * 4) * DIM + d] = v;
}

// ---------------------------------------------------------------------------
// SpMM: side[r] += vals[e] * ego[c]   (one wave32 per edge, 2 dims per lane)
// ego read from out[:, hop, :] (row = 256B contiguous; table fits in L2)
// ---------------------------------------------------------------------------
__global__ __launch_bounds__(256) void ngcf_spmm(const float* __restrict__ vals,
                                                 const int* __restrict__ rows,
                                                 const int* __restrict__ cols,
                                                 const float* __restrict__ outbuf,
                                                 float* __restrict__ side,
                                                 int hop) {
  int wave = (blockIdx.x * blockDim.x + threadIdx.x) >> 5;
  if (wave >= NNZ) return;
  int lane = threadIdx.x & 31;
  float v = vals[wave];
  int r = rows[wave];
  int c = cols[wave];
  const float* src = outbuf + ((size_t)c * 4 + hop) * DIM + lane * 2;
  float2 g = *(const float2*)src;
  float* dst = side + (size_t)r * DIM + lane * 2;
  unsafeAtomicAdd(dst + 0, v * g.x);   // global_atomic_add_f32
  unsafeAtomicAdd(dst + 1, v * g.y);
}

// ---------------------------------------------------------------------------
// Fused transform for one hop, one wave per 16-node tile:
//   acc = side @ Wgc + (ego*side) @ Wbi   (V_WMMA_F32_16X16X4_F32, full f32)
//   acc = leaky_relu(acc + bgc + bbi, 0.2)
//   out[:, hop+1, :] = acc / max(||row||, 1e-12)
//
// Weights are staged TRANSPOSED into LDS once per block (shared by 8 waves):
// sW[c*66 + k] = W[k][c], so a B operand {W[k0][col], W[k0+1][col]} is one
// 8B-aligned ds_load_b64. Stride 66 avoids the 64-way bank conflict a
// stride-64 transpose would cause.
//
// A 16x4 f32 layout: M = lane%16; lanes 0-15 hold K={0,1}, lanes 16-31 K={2,3}.
// C/D layout (ISA 7.12.2): lane L, VGPR r -> (M = r + 8*(L>=16), N = L%16).
// ---------------------------------------------------------------------------
__global__ __launch_bounds__(256) void ngcf_transform(
    const float* __restrict__ side, float* __restrict__ outbuf,
    const float* __restrict__ Wgc, const float* __restrict__ bgc,
    const float* __restrict__ Wbi, const float* __restrict__ bbi, int hop) {
  __shared__ float sWgc[DIM * LDS_STRIDE];
  __shared__ float sWbi[DIM * LDS_STRIDE];
  __shared__ float sBias[DIM];

  // Stage weights (transposed) + combined bias; all waves participate.
  for (int idx = threadIdx.x; idx < DIM * DIM; idx += 256) {
    int r = idx >> 6;        // K row
    int c = idx & 63;        // N col
    sWgc[c * LDS_STRIDE + r] = Wgc[idx];
    sWbi[c * LDS_STRIDE + r] = Wbi[idx];
  }
  if (threadIdx.x < DIM) sBias[threadIdx.x] = bgc[threadIdx.x] + bbi[threadIdx.x];
  __syncthreads();

  int wave = (blockIdx.x * blockDim.x + threadIdx.x) >> 5;
  if (wave >= N_TILES) return;
  int lane = threadIdx.x & 31;
  int h = lane >> 4;    // half-wave: 0 or 1
  int l16 = lane & 15;
  int rowBase = wave * 16;

  v8f acc[4] = {v8f{}, v8f{}, v8f{}, v8f{}};

  int m = rowBase + l16;
  const float* srow = side + (size_t)m * DIM;
  const float* erow = outbuf + ((size_t)m * 4 + hop) * DIM;

#pragma unroll 4
  for (int kk = 0; kk < 16; ++kk) {
    int k0 = kk * 4 + 2 * h;                 // this lane's K base
    v2f a_s = *(const v2f*)(srow + k0);      // global_load_b64
    v2f e2  = *(const v2f*)(erow + k0);
    v2f a_b;
    a_b.x = e2.x * a_s.x;                    // ego (*) side Hadamard
    a_b.y = e2.y * a_s.y;
#pragma unroll
    for (int n = 0; n < 4; ++n) {
      int col = n * 16 + l16;
      v2f bg = *(const v2f*)&sWgc[col * LDS_STRIDE + k0];  // ds_load_b64
      v2f bb = *(const v2f*)&sWbi[col * LDS_STRIDE + k0];
      acc[n] = __builtin_amdgcn_wmma_f32_16x16x4_f32(false, a_s, false, bg,
                                                     (short)0, acc[n], false, false);
      acc[n] = __builtin_amdgcn_wmma_f32_16x16x4_f32(false, a_b, false, bb,
                                                     (short)0, acc[n], false, false);
    }
  }

  // Bias + leaky ReLU + per-row sum of squares (row M = r + 8*h).
  float sumsq[8];
#pragma unroll
  for (int r = 0; r < 8; ++r) sumsq[r] = 0.0f;
#pragma unroll
  for (int n = 0; n < 4; ++n) {
    int col = n * 16 + l16;
    float bias = sBias[col];
#pragma unroll
    for (int r = 0; r < 8; ++r) {
      float v = acc[n][r] + bias;
      v = (v >= 0.0f) ? v : 0.2f * v;
      acc[n][r] = v;
      sumsq[r] += v * v;
    }
  }
  // Row values live across the 16 lanes of one half-wave: butterfly reduce.
#pragma unroll
  for (int mask = 1; mask <= 8; mask <<= 1) {
#pragma unroll
    for (int r = 0; r < 8; ++r) sumsq[r] += __shfl_xor(sumsq[r], mask, 32);
  }
  float scale[8];
#pragma unroll
  for (int r = 0; r < 8; ++r) {
    float nrm = sqrtf(sumsq[r]);
    scale[r] = 1.0f / fmaxf(nrm, 1e-12f);
  }
  // Normalized store into out[:, hop+1, :] (lane-consecutive cols -> coalesced).
#pragma unroll
  for (int r = 0; r < 8; ++r) {
    int node = rowBase + r + 8 * h;
    float* orow = outbuf + ((size_t)node * 4 + hop + 1) * DIM;
#pragma unroll
    for (int n = 0; n < 4; ++n) {
      int col = n * 16 + l16;
      orow[col] = acc[n][r] * scale[r];
    }
  }
}

// ---------------------------------------------------------------------------
extern "C" void kernel_launch(void* const* d_in, const int* in_sizes, int n_in,
                              void* d_out, int out_size, void* d_ws, size_t ws_size,
                              hipStream_t stream) {
  const float* user = (const float*)d_in[0];
  const float* item = (const float*)d_in[1];
  const float* Wgc  = (const float*)d_in[2];  // [3,64,64]
  const float* bgc  = (const float*)d_in[3];  // [3,1,64]
  const float* Wbi  = (const float*)d_in[4];
  const float* bbi  = (const float*)d_in[5];
  const float* vals = (const float*)d_in[6];
  const int* rows   = (const int*)d_in[7];
  const int* cols   = (const int*)d_in[8];
  float* out = (float*)d_out;
  float* side = (float*)d_ws;  // [N_NODES, 64] f32 = 38.4 MB scratch

  (void)in_sizes; (void)n_in; (void)out_size; (void)ws_size;

  ngcf_init<<<(N_NODES * DIM + 255) / 256, 256, 0, stream>>>(user, item, out);

  for (int k = 0; k < HOPS; ++k) {
    hipMemsetAsync(side, 0, (size_t)N_NODES * DIM * sizeof(float), stream);
    // one wave32 per edge
    long long spmm_threads = (long long)NNZ * 32;
    ngcf_spmm<<<(int)((spmm_threads + 255) / 256), 256, 0, stream>>>(
        vals, rows, cols, out, side, k);
    // one wave32 per 16-node tile, 8 waves per block
    ngcf_transform<<<(N_TILES + 7) / 8, 256, 0, stream>>>(
        side, out, Wgc + k * DIM * DIM, bgc + k * DIM, Wbi + k * DIM * DIM,
        bbi + k * DIM, k);
  }
}